// TypeGraphConvolution_29618094473318
// MI455X (gfx1250) — compile-verified
//
#include <hip/hip_runtime.h>

// B=8, L=128, D=768
#define BB 8
#define LL 128
#define DD 768

typedef __attribute__((ext_vector_type(2))) float v2f;
typedef __attribute__((ext_vector_type(4))) float v4f;
typedef __attribute__((ext_vector_type(8))) float v8f;

// ---------------------------------------------------------------------------
// Kernel 1: agg[b,i,d] = sum_j adj[b,i,j] * (text[b,j,d] + dep[b,j,i,d])
// One block per (b,i). 192 threads x float4 = 768 floats per row.
// dep_embed is 403 MB streamed exactly once -> non-temporal loads (TH=NT)
// so it does not evict text/W/agg from L2.
// ---------------------------------------------------------------------------
__global__ __launch_bounds__(192) void tgc_aggregate(
    const float* __restrict__ text,
    const float* __restrict__ adj,
    const float* __restrict__ dep,
    float* __restrict__ agg) {
  __shared__ float sAdj[LL];

  const int bi = blockIdx.x;          // 0..1023
  const int b  = bi >> 7;
  const int i  = bi & (LL - 1);
  const int t  = threadIdx.x;         // 0..191

  if (t < LL) sAdj[t] = adj[((size_t)b * LL + i) * LL + t];
  __syncthreads();

  const int d0 = t * 4;
  const float* tp = text + (size_t)b * LL * DD + d0;                    // text[b,0,d0]
  const float* dp = dep + (((size_t)b * LL) * LL + i) * (size_t)DD + d0; // dep[b,0,i,d0]

  v4f acc = {0.f, 0.f, 0.f, 0.f};
#pragma unroll 4
  for (int j = 0; j < LL; ++j) {
    const float w = sAdj[j];
    v4f tv = *(const v4f*)tp;                                // cached (text reused per i)
    v4f ev = __builtin_nontemporal_load((const v4f*)dp);     // streamed once
    acc.x += w * (tv.x + ev.x);
    acc.y += w * (tv.y + ev.y);
    acc.z += w * (tv.z + ev.z);
    acc.w += w * (tv.w + ev.w);
    tp += DD;
    dp += (size_t)LL * DD;
  }
  *(v4f*)(agg + (size_t)bi * DD + d0) = acc;
}

// ---------------------------------------------------------------------------
// Kernel 2: out[m, n] = relu(agg[m, :] @ W[:, n] + bias[n]),  M=1024, N=K=768
// Full-precision fp32 WMMA: V_WMMA_F32_16X16X4_F32.
// One wave -> 16x64 output tile (4 accumulators share one A fragment).
// 4 waves per block cover 64 rows x 64 cols; all waves in a block share the
// same B columns, so W fragments hit L0/L2.
//
// A 16x4 f32 layout : lanes 0-15 -> M=lane, K={0,1}; lanes 16-31 -> K={2,3}
// B  4x16 f32 layout: lanes 0-15 -> N=lane, K={0,1}; lanes 16-31 -> K={2,3}
// C/D 16x16 f32     : VGPR r -> M = r + 8*(lane>=16), N = lane&15
// ---------------------------------------------------------------------------
__global__ __launch_bounds__(128) void tgc_gemm_bias_relu(
    const float* __restrict__ agg,
    const float* __restrict__ W,      // W[d][e], row-major, d = K, e = N
    const float* __restrict__ bias,
    float* __restrict__ out) {
  const int wave   = threadIdx.x >> 5;   // 0..3
  const int lane   = threadIdx.x & 31;
  const int laneLo = lane & 15;
  const int laneHi = lane >> 4;          // 0 or 1

  const int nBase = blockIdx.x * 64;             // 12 n-tiles
  const int mTile = blockIdx.y * 4 + wave;       // 64 m-tiles
  const int row   = mTile * 16 + laneLo;         // A-matrix row for this lane

  v8f acc0 = {}, acc1 = {}, acc2 = {}, acc3 = {};

  const float* aRow = agg + (size_t)row * DD;
  for (int k = 0; k < DD; k += 4) {
    const int kk = k + laneHi * 2;
    // A fragment: two consecutive K values -> single b64 load
    v2f a = *(const v2f*)(aRow + kk);

    const float* wp = W + (size_t)kk * DD + nBase + laneLo;
    v2f b0, b1, b2, b3;
    b0.x = wp[0];   b0.y = wp[DD];
    b1.x = wp[16];  b1.y = wp[DD + 16];
    b2.x = wp[32];  b2.y = wp[DD + 32];
    b3.x = wp[48];  b3.y = wp[DD + 48];

    acc0 = __builtin_amdgcn_wmma_f32_16x16x4_f32(false, a, false, b0, (short)0, acc0, false, false);
    acc1 = __builtin_amdgcn_wmma_f32_16x16x4_f32(false, a, false, b1, (short)0, acc1, false, false);
    acc2 = __builtin_amdgcn_wmma_f32_16x16x4_f32(false, a, false, b2, (short)0, acc2, false, false);
    acc3 = __builtin_amdgcn_wmma_f32_16x16x4_f32(false, a, false, b3, (short)0, acc3, false, false);
  }

  // Epilogue: bias + relu, per C/D VGPR layout
  const int rowBase = mTile * 16 + laneHi * 8;
  const int col0    = nBase + laneLo;
  const float bs0 = bias[col0];
  const float bs1 = bias[col0 + 16];
  const float bs2 = bias[col0 + 32];
  const float bs3 = bias[col0 + 48];
#pragma unroll
  for (int e = 0; e < 8; ++e) {
    float* op = out + (size_t)(rowBase + e) * DD + col0;
    op[0]  = fmaxf(acc0[e] + bs0, 0.f);
    op[16] = fmaxf(acc1[e] + bs1, 0.f);
    op[32] = fmaxf(acc2[e] + bs2, 0.f);
    op[48] = fmaxf(acc3[e] + bs3, 0.f);
  }
}

extern "C" void kernel_launch(void* const* d_in, const int* in_sizes, int n_in,
                              void* d_out, int out_size, void* d_ws, size_t ws_size,
                              hipStream_t stream) {
  const float* text = (const float*)d_in[0];   // (B,L,D)
  const float* adj  = (const float*)d_in[1];   // (B,L,L)
  const float* dep  = (const float*)d_in[2];   // (B,L,L,D)
  const float* W    = (const float*)d_in[3];   // (D,D)
  const float* bias = (const float*)d_in[4];   // (D,)
  float* out = (float*)d_out;                  // (B,L,D)

  float* agg = (float*)d_ws;                   // needs B*L*D*4 = 3 MB scratch

  // Stage 1: streaming weighted aggregation (HBM-bound, 403 MB of dep_embed)
  tgc_aggregate<<<dim3(BB * LL), dim3(192), 0, stream>>>(text, adj, dep, agg);

  // Stage 2: fp32 WMMA GEMM + bias + relu. Grid: 12 n-tiles x 16 m-groups.
  tgc_gemm_bias_relu<<<dim3(DD / 64, (BB * LL) / 64), dim3(128), 0, stream>>>(agg, W, bias, out);
}